// PermutoEncoding_43671227466018
// MI455X (gfx1250) — compile-verified
//
#include <hip/hip_runtime.h>
#include <math.h>
#include <stdint.h>

#define NPTS 262144
#define NLEV 24
#define CAP  262144u
#define CAPM (CAP - 1u)
#define BLK  256

struct Params {
    float inv_scale[NLEV];
    float sf0, sf1, sf2;
};

typedef __attribute__((ext_vector_type(4))) unsigned int v4u;
typedef __attribute__((ext_vector_type(8))) int          v8i;
typedef __attribute__((ext_vector_type(4))) int          v4i;

__global__ __launch_bounds__(BLK) void permuto_kernel(
    const float* __restrict__ pos,      // (N, 3)
    const float* __restrict__ lat,      // (L, CAP, 2)
    const float* __restrict__ shift,    // (L, 3)
    const float* __restrict__ anneal,   // (L,)
    float* __restrict__ out,            // (N, L*2)
    Params P)
{
    __shared__ float smem[BLK * NLEV * 2];   // 48 KB staging (of 320 KB/WGP)

    const int tid = threadIdx.x;
    const int n   = blockIdx.x * BLK + tid;

    const float x0 = pos[3 * n + 0];
    const float x1 = pos[3 * n + 1];
    const float x2 = pos[3 * n + 2];

    #pragma unroll 2
    for (int l = 0; l < NLEV; ++l) {
        const float inv_s = P.inv_scale[l];
        // uniform per-level data -> scalar loads
        const float s0 = shift[3 * l + 0];
        const float s1 = shift[3 * l + 1];
        const float s2 = shift[3 * l + 2];
        const float aw = anneal[l];

        const float cf0 = (x0 * inv_s + s0) * P.sf0;
        const float cf1 = (x1 * inv_s + s1) * P.sf1;
        const float cf2 = (x2 * inv_s + s2) * P.sf2;

        // elevate: E * cf  (columns sum to zero)
        const float e0 = cf0 + cf1 + cf2;
        const float e1 = cf1 + cf2 - cf0;
        const float e2 = cf2 - 2.f * cf1;
        const float e3 = -3.f * cf2;

        // nearest remainder-0 lattice point (RNE == jnp.round)
        float r0 = rintf(e0 * 0.25f) * 4.f;
        float r1 = rintf(e1 * 0.25f) * 4.f;
        float r2 = rintf(e2 * 0.25f) * 4.f;
        float r3 = rintf(e3 * 0.25f) * 4.f;

        const float d0 = e0 - r0, d1 = e1 - r1, d2 = e2 - r2, d3 = e3 - r3;

        // descending rank with index tie-break: 6 pairwise compares
        int k0 = 0, k1 = 0, k2 = 0, k3 = 0;
        if (d0 < d1) k0++; else k1++;
        if (d0 < d2) k0++; else k2++;
        if (d0 < d3) k0++; else k3++;
        if (d1 < d2) k1++; else k2++;
        if (d1 < d3) k1++; else k3++;
        if (d2 < d3) k2++; else k3++;

        const int ss = (int)rintf((r0 + r1 + r2 + r3) * 0.25f);
        k0 += ss; k1 += ss; k2 += ss; k3 += ss;

        if (k0 < 0) { k0 += 4; r0 += 4.f; }  if (k0 > 3) { k0 -= 4; r0 -= 4.f; }
        if (k1 < 0) { k1 += 4; r1 += 4.f; }  if (k1 > 3) { k1 -= 4; r1 -= 4.f; }
        if (k2 < 0) { k2 += 4; r2 += 4.f; }  if (k2 > 3) { k2 -= 4; r2 -= 4.f; }
        if (k3 < 0) { k3 += 4; r3 += 4.f; }  if (k3 > 3) { k3 -= 4; r3 -= 4.f; }

        // hash the 4 simplex vertices (first 3 coords only), issue gathers early
        const int i0 = (int)r0, i1 = (int)r1, i2 = (int)r2;
        uint32_t idxv[4];
        #pragma unroll
        for (int v = 0; v < 4; ++v) {
            const int key0 = i0 + v - ((k0 > 3 - v) ? 4 : 0);
            const int key1 = i1 + v - ((k1 > 3 - v) ? 4 : 0);
            const int key2 = i2 + v - ((k2 > 3 - v) ? 4 : 0);
            const uint32_t h = (uint32_t)key0 * 2654435761u
                             ^ (uint32_t)key1 * 805459861u
                             ^ (uint32_t)key2 * 3674653429u;
            idxv[v] = h & CAPM;
        }
        const float2* tbl = (const float2*)lat + (size_t)l * CAP;
        const float2 g0v = tbl[idxv[0]];
        const float2 g1v = tbl[idxv[1]];
        const float2 g2v = tbl[idxv[2]];
        const float2 g3v = tbl[idxv[3]];

        // barycentric weights (computed while gathers are in flight)
        const float dl0 = (e0 - r0) * 0.25f;
        const float dl1 = (e1 - r1) * 0.25f;
        const float dl2 = (e2 - r2) * 0.25f;
        const float dl3 = (e3 - r3) * 0.25f;

        float b0 = 0.f, b1 = 0.f, b2 = 0.f, b3 = 0.f, b4 = 0.f;
        auto acc = [&](int k, float dl) {
            b0 += (k == 3) ? dl : 0.f;
            b1 += (k == 2) ? dl : 0.f;  b1 -= (k == 3) ? dl : 0.f;
            b2 += (k == 1) ? dl : 0.f;  b2 -= (k == 2) ? dl : 0.f;
            b3 += (k == 0) ? dl : 0.f;  b3 -= (k == 1) ? dl : 0.f;
            b4 -= (k == 0) ? dl : 0.f;
        };
        acc(k0, dl0); acc(k1, dl1); acc(k2, dl2); acc(k3, dl3);

        const float w0 = b0 + 1.f + b4;
        const float f0 = w0 * g0v.x + b1 * g1v.x + b2 * g2v.x + b3 * g3v.x;
        const float f1 = w0 * g0v.y + b1 * g1v.y + b2 * g2v.y + b3 * g3v.y;

        *(float2*)&smem[tid * (NLEV * 2) + l * 2] = make_float2(f0 * aw, f1 * aw);
    }

    __syncthreads();

#if __has_builtin(__builtin_amdgcn_tensor_store_from_lds)
    // One TDM descriptor per block: contiguous 48 KB LDS -> global DMA store.
    if (tid < 32) {   // wave 0 only (TDM ignores EXEC; must be branched around)
        const uint64_t ga = (uint64_t)(uintptr_t)out
                          + (uint64_t)blockIdx.x * (uint64_t)(BLK * NLEV * 2 * 4);
        const uint32_t lds_off = (uint32_t)(uintptr_t)&smem[0];
        const uint32_t elems   = (BLK * NLEV * 2) / 2;   // 6144 x 8-byte elements

        v4u g0;
        g0[0] = 1u;                                              // count=1, user mode
        g0[1] = lds_off;                                         // lds_addr
        g0[2] = (uint32_t)ga;                                    // global_addr[31:0]
        g0[3] = (uint32_t)((ga >> 32) & 0x1FFFFFFull) | (2u << 30); // addr[56:32] | type=2

        v8i g1;
        g1[0] = (int)(3u << 16);                 // wg_mask=0, data_size=3 (8B), no opts
        g1[1] = (int)((elems & 0xFFFFu) << 16);  // tensor_dim0[15:0]
        g1[2] = (int)((elems >> 16) | (1u << 16)); // tensor_dim0[31:16] | tensor_dim1=1
        g1[3] = (int)(elems << 16);              // tensor_dim1[31:16]=0 | tile_dim0
        g1[4] = 0;                               // tile_dim1=0, tile_dim2=0
        g1[5] = (int)elems;                      // tensor_dim0_stride[31:0]
        g1[6] = 0;                               // stride hi | tensor_dim1_stride lo
        g1[7] = 0;

        v4i gz4 = {0, 0, 0, 0};                  // groups 2/3 unused (<=2D tensor)
        v8i gz8 = {0, 0, 0, 0, 0, 0, 0, 0};      // trailing group (unused, zero)
        __builtin_amdgcn_tensor_store_from_lds(g0, g1, gz4, gz4, gz8, 0);
  #if __has_builtin(__builtin_amdgcn_s_wait_tensorcnt)
        __builtin_amdgcn_s_wait_tensorcnt(0);
  #endif
    }
#else
    // Fallback: cooperative fully-coalesced copy LDS -> global
    const float4* s4 = (const float4*)smem;
    float4* o4 = (float4*)out + (size_t)blockIdx.x * ((BLK * NLEV * 2) / 4);
    #pragma unroll
    for (int j = 0; j < (NLEV * 2) / 4; ++j)
        o4[tid + j * BLK] = s4[tid + j * BLK];
#endif
}

extern "C" void kernel_launch(void* const* d_in, const int* in_sizes, int n_in,
                              void* d_out, int out_size, void* d_ws, size_t ws_size,
                              hipStream_t stream) {
    (void)in_sizes; (void)n_in; (void)out_size; (void)d_ws; (void)ws_size;

    Params P;
    for (int i = 0; i < NLEV; ++i) {
        // scales = geomspace(1, 1e-4, 24): 10^(-4*i/23) in double, as in numpy
        const double s = pow(10.0, -4.0 * (double)i / 23.0);
        P.inv_scale[i] = (float)(1.0 / s);
    }
    const double inv_std = 4.0 * sqrt(2.0 / 3.0);
    P.sf0 = (float)(inv_std / sqrt(1.0 * 2.0));
    P.sf1 = (float)(inv_std / sqrt(2.0 * 3.0));
    P.sf2 = (float)(inv_std / sqrt(3.0 * 4.0));

    const float* pos    = (const float*)d_in[0];
    const float* lat    = (const float*)d_in[1];
    const float* shift  = (const float*)d_in[2];
    const float* anneal = (const float*)d_in[3];
    float* out          = (float*)d_out;

    permuto_kernel<<<NPTS / BLK, BLK, 0, stream>>>(pos, lat, shift, anneal, out, P);
}